// Corr2Cost_57312043598467
// MI455X (gfx1250) — compile-verified
//
#include <hip/hip_runtime.h>
#include <hip/hip_bf16.h>

#define AS1 __attribute__((address_space(1)))
#define AS3 __attribute__((address_space(3)))

typedef int v4i __attribute__((ext_vector_type(4)));

namespace {
constexpr int Bn = 8, Cn = 2, Dn = 128, Hn = 96, Wn = 128;
constexpr int THREADS = 256;   // 8 waves (wave32)
}

// Issue one async 16-byte global->LDS copy (CDNA5 GLOBAL_LOAD_ASYNC_TO_LDS_B128).
__device__ __forceinline__ void async_copy_b128(const float* g, AS3 char* l) {
#if __has_builtin(__builtin_amdgcn_global_load_async_to_lds_b128)
  __builtin_amdgcn_global_load_async_to_lds_b128(
      (AS1 v4i*)g, (AS3 v4i*)l, /*offset=*/0, /*cpol=*/0);
#else
  unsigned      lds_addr = (unsigned)(unsigned long long)l;       // LDS byte offset
  unsigned long long ga  = (unsigned long long)g;                 // 64-bit global addr (GV mode)
  asm volatile("global_load_async_to_lds_b128 %0, %1, off"
               :: "v"(lds_addr), "v"(ga) : "memory");
#endif
}

__device__ __forceinline__ void wait_asynccnt0() {
#if __has_builtin(__builtin_amdgcn_s_wait_asynccnt)
  __builtin_amdgcn_s_wait_asynccnt(0);
#else
  asm volatile("s_wait_asynccnt 0" ::: "memory");
#endif
}

// With K = 2*maxdisp+1, jnp.linspace(-m, m, K) has step exactly 1.0f, so every
// sample position is an exact integer float: floor(pos)==pos, lerp weight w1==0.
// The reference therefore reduces *exactly* to a zero-padded integer gather:
//   out[b,c,k,i,j] = (0 <= base+k-m < D) ? corr[b,c, base+k-m, i, j] : 0
// where base = j (is_ux) or i (else).
__global__ void __launch_bounds__(THREADS)
corr2cost_kernel(const float* __restrict__ corr,
                 const int* __restrict__ p_maxdisp,
                 const int* __restrict__ p_isux,
                 float* __restrict__ out) {
  extern __shared__ float tile[];                   // Dn*Wn floats (64 KB)

  const int maxdisp = p_maxdisp[0];
  const int is_ux   = p_isux[0];
  const int K       = 2 * maxdisp + 1;

  const int  s = blockIdx.x / Hn;                   // s = b*C + c
  const int  i = blockIdx.x - s * Hn;               // H row handled by this block
  const long planeHW = (long)Hn * Wn;
  const float* slab  = corr + (long)s * Dn * planeHW + (long)i * Wn;  // row d: slab + d*planeHW
  float*       obase = out  + (long)s * K  * planeHW + (long)i * Wn;  // row k: obase + k*planeHW

  constexpr int NJ4 = Wn / 4;                       // float4 packs per row

  if (is_ux) {
    // ---- Stage D x W tile into LDS with async b128 copies (1 instr = 1 row/wave) ----
    const int wave = threadIdx.x >> 5;
    const int lane = threadIdx.x & 31;
    AS3 char* lbase = (AS3 char*)tile;
    #pragma unroll 4
    for (int d = wave; d < Dn; d += (THREADS >> 5)) {
      const float* g = slab + (long)d * planeHW + lane * 4;           // 16 B per lane
      async_copy_b128(g, lbase + (((long)d * Wn + lane * 4) << 2));
    }
    wait_asynccnt0();          // my wave's async copies have landed in LDS
    __syncthreads();           // everyone's copies visible

    // ---- Emit outputs: 4 diagonal LDS reads + one b128 store per thread-iter ----
    for (int idx = threadIdx.x; idx < K * NJ4; idx += THREADS) {
      const int j  = (idx % NJ4) * 4;
      const int k  = idx / NJ4;
      const int db = j + k - maxdisp;               // d index for element j
      const int a  = db * Wn + j;                   // LDS dword index; +t*(Wn+1) per element
      float4 v;
      v.x = ((unsigned)(db + 0) < (unsigned)Dn) ? tile[a]                : 0.0f;
      v.y = ((unsigned)(db + 1) < (unsigned)Dn) ? tile[a + 1*(Wn + 1)]   : 0.0f;
      v.z = ((unsigned)(db + 2) < (unsigned)Dn) ? tile[a + 2*(Wn + 1)]   : 0.0f;
      v.w = ((unsigned)(db + 3) < (unsigned)Dn) ? tile[a + 3*(Wn + 1)]   : 0.0f;
      *(float4*)(obase + (long)k * planeHW + j) = v;
    }
  } else {
    // Gather along H axis: tap depends only on k -> whole row copy (or zero), b128 both ways.
    for (int idx = threadIdx.x; idx < K * NJ4; idx += THREADS) {
      const int j = (idx % NJ4) * 4;
      const int k = idx / NJ4;
      const int d = i + k - maxdisp;
      float4 v = make_float4(0.0f, 0.0f, 0.0f, 0.0f);
      if ((unsigned)d < (unsigned)Dn)
        v = *(const float4*)(slab + (long)d * planeHW + j);
      *(float4*)(obase + (long)k * planeHW + j) = v;
    }
  }
}

extern "C" void kernel_launch(void* const* d_in, const int* in_sizes, int n_in,
                              void* d_out, int out_size, void* d_ws, size_t ws_size,
                              hipStream_t stream) {
  const float* corr    = (const float*)d_in[0];
  const int*   maxdisp = (const int*)d_in[1];
  const int*   isux    = (const int*)d_in[2];
  float*       out     = (float*)d_out;

  const dim3   grid(Bn * Cn * Hn);                       // 1536 blocks, one per (b,c,i)
  const size_t shmem = (size_t)Dn * Wn * sizeof(float);  // 64 KB tile in LDS
  corr2cost_kernel<<<grid, THREADS, shmem, stream>>>(corr, maxdisp, isux, out);
}